// Mamba2_14242111553808
// MI455X (gfx1250) — compile-verified
//
#include <hip/hip_runtime.h>
#include <hip/hip_bf16.h>
#include <math.h>

// ---------------- problem constants (fixed by reference) ----------------
#define BB        16
#define HH        56
#define WW        56
#define LL        (HH*WW)            // 3136
#define DMODEL    384
#define DINNER    768
#define HEADDIM   64
#define NHEADS    12
#define DSTATE    64
#define CONVDIM   (DINNER + 2*DSTATE)            // 896
#define DINPROJ   (2*DINNER + 2*DSTATE + NHEADS) // 1676
#define LN_EPS    1e-5f
#define NBL       ((size_t)BB*LL)    // 50176 rows

typedef __attribute__((ext_vector_type(2))) float v2f;
typedef __attribute__((ext_vector_type(8))) float v8f;

static __device__ __forceinline__ v8f wmma4(v2f a, v2f b, v8f c) {
    // D = A(16x4) * B(4x16) + C   (fp32 matrix pipe, exact fp32 accumulate)
    return __builtin_amdgcn_wmma_f32_16x16x4_f32(
        /*neg_a=*/false, a, /*neg_b=*/false, b,
        /*c_mod=*/(short)0, c, /*reuse_a=*/false, /*reuse_b=*/false);
}

static __device__ __forceinline__ v2f ld2(const float* p) {
    return *(const v2f*)p;  // 8B-aligned by construction (even float offsets)
}

static __device__ __forceinline__ float siluf(float x) {
    return x / (1.0f + expf(-x));
}

// ============================================================================
// 1/6) C[M][N] = A[M][K] @ W[N][K]^T    (M % 64 == 0, K % 16 == 0, N arbitrary)
//    block: 256 threads = 8 waves in a 2(M) x 4(N) grid; wave tile 32x64.
//    Branch-free inner loop: out-of-range B columns are CLAMPED to N-1 (their
//    accumulator columns are never stored), so no per-iteration predication.
// ============================================================================
__global__ void gemm_wmma_nt(const float* __restrict__ A,
                             const float* __restrict__ Wt,
                             float* __restrict__ C,
                             int M, int N, int K) {
    const int lane  = threadIdx.x & 31;
    const int wave  = threadIdx.x >> 5;
    const int lrow  = lane & 15;
    const int khalf = lane >> 4;          // 0: K pair {0,1}, 1: K pair {2,3}

    const int baseM = blockIdx.y * 64 + (wave & 1) * 32;
    const int baseN = blockIdx.x * 256 + (wave >> 1) * 64;

    // loop-invariant per-lane base pointers (indexed by k0 inside the loop)
    const float* a0p = A + (size_t)(baseM + lrow) * K + 2 * khalf;
    const float* a1p = a0p + (size_t)16 * K;
    const float* bp[4];
    #pragma unroll
    for (int nt = 0; nt < 4; ++nt) {
        int col = baseN + nt * 16 + lrow;
        col = (col < N) ? col : (N - 1);   // clamp: load valid, never stored
        bp[nt] = Wt + (size_t)col * K + 2 * khalf;
    }

    v8f acc[2][4];
    #pragma unroll
    for (int i = 0; i < 2; ++i)
        #pragma unroll
        for (int j = 0; j < 4; ++j) acc[i][j] = (v8f)0.0f;

    #pragma unroll 4
    for (int k0 = 0; k0 < K; k0 += 4) {
        v2f a0 = ld2(a0p + k0);
        v2f a1 = ld2(a1p + k0);
        v2f bf[4];
        #pragma unroll
        for (int nt = 0; nt < 4; ++nt) bf[nt] = ld2(bp[nt] + k0);
        #pragma unroll
        for (int nt = 0; nt < 4; ++nt) {
            acc[0][nt] = wmma4(a0, bf[nt], acc[0][nt]);
            acc[1][nt] = wmma4(a1, bf[nt], acc[1][nt]);
        }
    }

    #pragma unroll
    for (int mi = 0; mi < 2; ++mi) {
        #pragma unroll
        for (int nt = 0; nt < 4; ++nt) {
            const int col = baseN + nt * 16 + lrow;
            if (col < N) {
                #pragma unroll
                for (int r = 0; r < 8; ++r) {
                    const int row = baseM + mi * 16 + khalf * 8 + r;
                    C[(size_t)row * N + col] = acc[mi][nt][r];
                }
            }
        }
    }
}

// ============================================================================
// 2) dtA[bl][h] = softplus(dt_raw + dt_bias[h]) * (-exp(A_log[h]))
// ============================================================================
__global__ void dtA_kernel(const float* __restrict__ zxbcdt,
                           const float* __restrict__ dt_bias,
                           const float* __restrict__ A_log,
                           float* __restrict__ dtA) {
    const size_t idx = (size_t)blockIdx.x * blockDim.x + threadIdx.x;
    if (idx >= NBL * NHEADS) return;
    const int h  = (int)(idx % NHEADS);
    const size_t bl = idx / NHEADS;
    float x = zxbcdt[bl * DINPROJ + (2 * DINNER + 2 * DSTATE) + h] + dt_bias[h];
    float sp = (x > 20.0f) ? x : log1pf(expf(x));
    dtA[idx] = sp * (-expf(A_log[h]));
}

// ============================================================================
// 3) depthwise 3x3 conv (pad 1) + bias + SiLU over xBC slice of zxbcdt
//    channel-innermost layout -> 32 lanes read contiguous 128B per tap
// ============================================================================
__global__ void conv_dw_kernel(const float* __restrict__ zxbcdt,
                               const float* __restrict__ conv_w,
                               const float* __restrict__ conv_b,
                               float* __restrict__ xconv) {
    const size_t idx = (size_t)blockIdx.x * blockDim.x + threadIdx.x;
    if (idx >= NBL * CONVDIM) return;
    const int c = (int)(idx % CONVDIM);
    const size_t t = idx / CONVDIM;
    const int l = (int)(t % LL);
    const int b = (int)(t / LL);
    const int y = l / WW, x = l % WW;

    const float* wgt = conv_w + (size_t)c * 9;
    float acc = conv_b[c];
    #pragma unroll
    for (int ky = -1; ky <= 1; ++ky) {
        const int yy = y + ky;
        if (yy < 0 || yy >= HH) continue;
        #pragma unroll
        for (int kx = -1; kx <= 1; ++kx) {
            const int xx = x + kx;
            if (xx < 0 || xx >= WW) continue;
            const size_t ll = (size_t)b * LL + yy * WW + xx;
            acc += wgt[(ky + 1) * 3 + (kx + 1)] *
                   zxbcdt[ll * DINPROJ + DINNER + c];
        }
    }
    xconv[idx] = siluf(acc);
}

// ============================================================================
// 4) KV[b][h][s][p] = sum_l Bmat[b][l][s] * (x[b][l][h*64+p] * dtA[b][l][h])
//    one block per (b,h); 8 waves, each owns 2 of 16 output 16x16 tiles,
//    full K = 3136 loop with hoisted per-lane base pointers.
// ============================================================================
__global__ void kv_kernel(const float* __restrict__ xconv,
                          const float* __restrict__ dtA,
                          float* __restrict__ KV) {
    const int bh = blockIdx.x;               // 0 .. B*NHEADS-1
    const int b  = bh / NHEADS;
    const int h  = bh % NHEADS;

    const int lane  = threadIdx.x & 31;
    const int wave  = threadIdx.x >> 5;
    const int lrow  = lane & 15;
    const int khalf = lane >> 4;

    const int mt  = wave & 3;                // s-tile (shared by both units)
    const int nt0 = wave >> 2;               // p-tiles nt0 and nt0+2
    const int nt1 = nt0 + 2;

    // hoisted per-lane base pointers at l = 2*khalf
    const float* base = xconv + (size_t)b * LL * CONVDIM
                        + (size_t)(2 * khalf) * CONVDIM;
    const float* sp0  = base + DINNER + mt * 16 + lrow;          // Bmat[s]
    const float* pp0  = base + h * HEADDIM + nt0 * 16 + lrow;    // x[.. p0]
    const float* pp1  = base + h * HEADDIM + nt1 * 16 + lrow;    // x[.. p1]
    const float* dtp  = dtA + (size_t)b * LL * NHEADS + (2 * khalf) * NHEADS + h;

    v8f acc0 = (v8f)0.0f, acc1 = (v8f)0.0f;

    #pragma unroll 2
    for (int l0 = 0; l0 < LL; l0 += 4) {
        const size_t o0 = (size_t)l0 * CONVDIM;
        const size_t o1 = o0 + CONVDIM;
        const float dA0 = dtp[(size_t)l0 * NHEADS];
        const float dA1 = dtp[(size_t)(l0 + 1) * NHEADS];

        v2f a;  a.x  = sp0[o0];          a.y  = sp0[o1];
        v2f b0; b0.x = pp0[o0] * dA0;    b0.y = pp0[o1] * dA1;
        v2f b1; b1.x = pp1[o0] * dA0;    b1.y = pp1[o1] * dA1;

        acc0 = wmma4(a, b0, acc0);
        acc1 = wmma4(a, b1, acc1);
    }

    float* out = KV + (size_t)bh * DSTATE * HEADDIM;
    #pragma unroll
    for (int r = 0; r < 8; ++r) {
        const int row = mt * 16 + khalf * 8 + r;
        out[row * HEADDIM + nt0 * 16 + lrow] = acc0[r];
        out[row * HEADDIM + nt1 * 16 + lrow] = acc1[r];
    }
}

// ============================================================================
// 5) y = Cmat @ KV + V*D ; y *= silu(z) ; LayerNorm(768) -> yn
//    one block per (b, 16-row l tile); y tile staged in 48KB LDS.
// ============================================================================
__global__ void y_ln_kernel(const float* __restrict__ xconv,
                            const float* __restrict__ KV,
                            const float* __restrict__ zxbcdt,
                            const float* __restrict__ Dp,
                            const float* __restrict__ ln_g,
                            const float* __restrict__ ln_b,
                            float* __restrict__ yn) {
    __shared__ float yt[16][DINNER];         // 49152 bytes

    const int blk = blockIdx.x;
    const int b   = blk / (LL / 16);
    const int lt  = (blk % (LL / 16)) * 16;

    const int lane  = threadIdx.x & 31;
    const int wave  = threadIdx.x >> 5;
    const int lrow  = lane & 15;
    const int khalf = lane >> 4;

    // --- phase 1: 48 (head, 16-col tile) units over 8 waves, WMMA K=64 ---
    for (int u = wave; u < NHEADS * 4; u += 8) {
        const int h  = u >> 2;
        const int nt = u & 3;
        const int pc = h * HEADDIM + nt * 16 + lrow;   // global channel col
        const float dcoef = Dp[h];

        // init accumulator with skip term V*D in WMMA C layout
        v8f acc;
        #pragma unroll
        for (int r = 0; r < 8; ++r) {
            const int l = lt + khalf * 8 + r;
            acc[r] = xconv[((size_t)b * LL + l) * CONVDIM + pc] * dcoef;
        }

        const float* kvp = KV + ((size_t)b * NHEADS + h) * DSTATE * HEADDIM
                           + nt * 16 + lrow;
        const int l = lt + lrow;
        const float* crow = xconv + ((size_t)b * LL + l) * CONVDIM
                            + (DINNER + DSTATE) + 2 * khalf;  // Cmat @ 832
        #pragma unroll
        for (int s0 = 0; s0 < DSTATE; s0 += 4) {
            v2f a = ld2(crow + s0);                    // contiguous pair in s
            v2f bf;
            bf.x = kvp[(s0 + 2 * khalf)     * HEADDIM];
            bf.y = kvp[(s0 + 2 * khalf + 1) * HEADDIM];
            acc = wmma4(a, bf, acc);
        }

        #pragma unroll
        for (int r = 0; r < 8; ++r)
            yt[khalf * 8 + r][pc] = acc[r];
    }
    __syncthreads();

    // --- phase 2: gating + LayerNorm, 2 rows per wave ---
    #pragma unroll
    for (int rr = 0; rr < 2; ++rr) {
        const int row = wave * 2 + rr;
        const int l   = lt + row;
        const float* zrow = zxbcdt + ((size_t)b * LL + l) * DINPROJ; // z first
        float s1 = 0.0f, s2 = 0.0f;
        #pragma unroll 4
        for (int c = lane; c < DINNER; c += 32) {
            float yg = yt[row][c] * siluf(zrow[c]);
            yt[row][c] = yg;
            s1 += yg;
            s2 += yg * yg;
        }
        #pragma unroll
        for (int off = 16; off > 0; off >>= 1) {
            s1 += __shfl_xor(s1, off);
            s2 += __shfl_xor(s2, off);
        }
        const float mu  = s1 * (1.0f / DINNER);
        const float var = s2 * (1.0f / DINNER) - mu * mu;
        const float inv = rsqrtf(var + LN_EPS);
        float* out = yn + ((size_t)b * LL + l) * DINNER;
        #pragma unroll 4
        for (int c = lane; c < DINNER; c += 32)
            out[c] = (yt[row][c] - mu) * inv * ln_g[c] + ln_b[c];
    }
}

// ============================================================================
// launcher
// ============================================================================
extern "C" void kernel_launch(void* const* d_in, const int* in_sizes, int n_in,
                              void* d_out, int out_size, void* d_ws, size_t ws_size,
                              hipStream_t stream) {
    const float* u       = (const float*)d_in[0];
    const float* W_in    = (const float*)d_in[1];
    const float* conv_w  = (const float*)d_in[2];
    const float* conv_b  = (const float*)d_in[3];
    const float* dt_bias = (const float*)d_in[4];
    const float* A_log   = (const float*)d_in[5];
    const float* Dp      = (const float*)d_in[6];
    const float* ln_g    = (const float*)d_in[7];
    const float* ln_b    = (const float*)d_in[8];
    const float* W_out   = (const float*)d_in[9];
    float* out = (float*)d_out;

    // workspace carving (floats)
    float* ws = (float*)d_ws;
    size_t off = 0;
    float* zxbcdt = ws + off; off += NBL * DINPROJ;          // 336 MB
    float* xconv  = ws + off; off += NBL * CONVDIM;          // 180 MB
    float* dtA    = ws + off; off += NBL * NHEADS;           // 2.4 MB
    float* KV     = ws + off; off += (size_t)BB * NHEADS * DSTATE * HEADDIM;
    float* yn     = ws + off; off += NBL * DINNER;           // 154 MB

    // 1) in_proj: zxbcdt = u @ W_in^T   (M=50176, N=1676, K=384)
    {
        dim3 grid((DINPROJ + 255) / 256, (int)(NBL / 64));
        gemm_wmma_nt<<<grid, 256, 0, stream>>>(u, W_in, zxbcdt,
                                               (int)NBL, DINPROJ, DMODEL);
    }
    // 2) dtA
    {
        const size_t n = NBL * NHEADS;
        dtA_kernel<<<(unsigned)((n + 255) / 256), 256, 0, stream>>>(
            zxbcdt, dt_bias, A_log, dtA);
    }
    // 3) depthwise conv + SiLU
    {
        const size_t n = NBL * CONVDIM;
        conv_dw_kernel<<<(unsigned)((n + 255) / 256), 256, 0, stream>>>(
            zxbcdt, conv_w, conv_b, xconv);
    }
    // 4) KV per (b,h)
    kv_kernel<<<BB * NHEADS, 256, 0, stream>>>(xconv, dtA, KV);
    // 5) y + gate + LN
    y_ln_kernel<<<BB * (LL / 16), 256, 0, stream>>>(xconv, KV, zxbcdt,
                                                    Dp, ln_g, ln_b, yn);
    // 6) out_proj: out = yn @ W_out^T  (M=50176, N=384, K=768)
    {
        dim3 grid((DMODEL + 255) / 256, (int)(NBL / 64));
        gemm_wmma_nt<<<grid, 256, 0, stream>>>(yn, W_out, out,
                                               (int)NBL, DMODEL, DINNER);
    }
    (void)in_sizes; (void)n_in; (void)out_size; (void)ws_size;
}